// RWKVRewardModel_46196668236242
// MI455X (gfx1250) — compile-verified
//
#include <hip/hip_runtime.h>
#include <hip/hip_bf16.h>
#include <math.h>

// ---------------- model dims ----------------
#define Lc  6
#define Cc  1024
#define HSc 64
#define Hc  16
#define Fc  3584
#define DWc 64
#define DAc 64
#define DVc 32
#define DGc 128
#define Bc  2
#define Tc  2048
#define BTc (Bc*Tc)

typedef __attribute__((ext_vector_type(2))) float v2f;
typedef __attribute__((ext_vector_type(8))) float v8f;

// -------- CDNA5 async Global->LDS copy (16B per lane, ASYNCcnt-tracked) ------
__device__ __forceinline__ void async_copy16(const float* gsrc, const float* ldst) {
    unsigned int laddr = (unsigned int)(unsigned long long)ldst;  // addr[31:0] = LDS offset
    asm volatile("global_load_async_to_lds_b128 %0, %1, off"
                 :: "v"(laddr), "v"(gsrc) : "memory");
}
__device__ __forceinline__ void async_wait0() {
    asm volatile("s_wait_asynccnt 0x0" ::: "memory");
}

// ---------------- LDS-tiled fp32 WMMA GEMM ----------------
// Block = 256 threads (8 waves) -> 64x64 output tile; K chunked by 32.
// Requires M%64==0, N%64==0, K%32==0. A 16x4 f32 WMMA layout:
// lanes 0-15 hold K=k,k+1 ; lanes 16-31 hold K=k+2,k+3. C/D: VGPR r = rows r/r+8.
__global__ void k_gemm_tiled(const float* __restrict__ X, const float* __restrict__ W,
                             float* __restrict__ Y, int M, int N, int K) {
    __shared__ float Xs[64 * 32];   // [row 0..63][k 0..31]
    __shared__ float Ws[32 * 64];   // [k 0..31][col 0..63]
    int tid  = threadIdx.x;
    int lane = tid & 31, wave = tid >> 5;
    int bn   = N >> 6;
    int m0   = (blockIdx.x / bn) * 64;
    int n0   = (blockIdx.x % bn) * 64;
    int wm   = wave >> 1;              // 0..3 : M subtile
    int wn2  = (wave & 1) * 2;         // 0 or 2: first of two N subtiles
    int half = lane >> 4, l = lane & 15;

    // cooperative load coordinates
    int xr = tid >> 3, xc = (tid & 7) << 2;    // X: 32 rows/pass, 4 floats each
    int wr_ = tid >> 4, wc = (tid & 15) << 2;  // W: 16 rows/pass, 4 floats each

    v8f acc0 = {0.f,0.f,0.f,0.f,0.f,0.f,0.f,0.f};
    v8f acc1 = {0.f,0.f,0.f,0.f,0.f,0.f,0.f,0.f};

    for (int k0 = 0; k0 < K; k0 += 32) {
        __syncthreads();  // previous chunk fully consumed before overwrite
        // stage X 64x32 (2 passes) and W 32x64 (2 passes) via async DMA
        async_copy16(X + (size_t)(m0 + xr) * K + k0 + xc,            &Xs[xr * 32 + xc]);
        async_copy16(X + (size_t)(m0 + 32 + xr) * K + k0 + xc,       &Xs[(32 + xr) * 32 + xc]);
        async_copy16(W + (size_t)(k0 + wr_) * N + n0 + wc,           &Ws[wr_ * 64 + wc]);
        async_copy16(W + (size_t)(k0 + 16 + wr_) * N + n0 + wc,      &Ws[(16 + wr_) * 64 + wc]);
        async_wait0();
        __syncthreads();
#pragma unroll
        for (int kk = 0; kk < 32; kk += 4) {
            int kb = kk + half * 2;
            v2f a = *(const v2f*)&Xs[(wm * 16 + l) * 32 + kb];
            v2f b0, b1;
            b0.x = Ws[kb * 64 + wn2 * 16 + l];
            b0.y = Ws[(kb + 1) * 64 + wn2 * 16 + l];
            b1.x = Ws[kb * 64 + (wn2 + 1) * 16 + l];
            b1.y = Ws[(kb + 1) * 64 + (wn2 + 1) * 16 + l];
            acc0 = __builtin_amdgcn_wmma_f32_16x16x4_f32(false, a, false, b0,
                                                         (short)0, acc0, false, false);
            acc1 = __builtin_amdgcn_wmma_f32_16x16x4_f32(false, a, false, b1,
                                                         (short)0, acc1, false, false);
        }
    }
    float* Y0 = Y + (size_t)(m0 + wm * 16 + half * 8) * N + n0 + wn2 * 16 + l;
#pragma unroll
    for (int r = 0; r < 8; ++r) { Y0[(size_t)r * N] = acc0[r]; Y0[(size_t)r * N + 16] = acc1[r]; }
}

// ---------------- fallback direct WMMA GEMM (for N=32 LoRA) ----------------
__global__ void k_gemm_f32(const float* __restrict__ X, const float* __restrict__ W,
                           float* __restrict__ Y, int M, int N, int K) {
    int lane = threadIdx.x & 31;
    int wave = blockIdx.x * (blockDim.x >> 5) + (threadIdx.x >> 5);
    int tilesN = N >> 4;
    int tiles  = (M >> 4) * tilesN;
    if (wave >= tiles) return;                 // wave-uniform
    int tm = wave / tilesN, tn = wave % tilesN;
    int half = lane >> 4;
    int l    = lane & 15;
    const float* Xrow = X + (size_t)(tm * 16 + l) * K;
    const float* Wcol = W + (tn * 16 + l);
    v8f c = {0.f,0.f,0.f,0.f,0.f,0.f,0.f,0.f};
    for (int k0 = 0; k0 < K; k0 += 4) {
        int kb = k0 + half * 2;
        v2f a = *(const v2f*)(Xrow + kb);
        v2f b;
        b.x = Wcol[(size_t)kb * N];
        b.y = Wcol[(size_t)(kb + 1) * N];
        c = __builtin_amdgcn_wmma_f32_16x16x4_f32(false, a, false, b,
                                                  (short)0, c, false, false);
    }
    float* Yt = Y + (size_t)(tm * 16 + half * 8) * N + tn * 16 + l;
#pragma unroll
    for (int r = 0; r < 8; ++r) Yt[(size_t)r * N] = c[r];
}

// ---------------- embedding gather ----------------
__global__ void k_embed(const float* __restrict__ emb, const int* __restrict__ idx,
                        float* __restrict__ out) {
    size_t i = (size_t)blockIdx.x * blockDim.x + threadIdx.x;
    if (i >= (size_t)BTc * Cc) return;
    size_t bt = i / Cc; int c = (int)(i % Cc);
    out[i] = emb[(size_t)idx[bt] * Cc + c];
}

// ---------------- per-token LayerNorm over C ----------------
__global__ void k_layernorm(const float* __restrict__ in, const float* __restrict__ w,
                            const float* __restrict__ b, float* __restrict__ out, float eps) {
    int bt = blockIdx.x, tid = threadIdx.x;
    __shared__ float s1[256], s2[256];
    const float* xp = in + (size_t)bt * Cc;
    float vals[4], ls = 0.f, lq = 0.f;
#pragma unroll
    for (int q = 0; q < 4; ++q) { float v = xp[tid * 4 + q]; vals[q] = v; ls += v; lq += v * v; }
    s1[tid] = ls; s2[tid] = lq; __syncthreads();
    for (int off = 128; off > 0; off >>= 1) {
        if (tid < off) { s1[tid] += s1[tid + off]; s2[tid] += s2[tid + off]; }
        __syncthreads();
    }
    float mean = s1[0] * (1.f / Cc);
    float var  = s2[0] * (1.f / Cc) - mean * mean;
    float rs   = rsqrtf(var + eps);
    float* op = out + (size_t)bt * Cc;
#pragma unroll
    for (int q = 0; q < 4; ++q) {
        int c = tid * 4 + q;
        op[c] = (vals[q] - mean) * rs * w[c] + b[c];
    }
}

// ---------------- token-shift mix: 6 outputs (r,w,k,v,a,g) ----------------
__global__ void k_mix6(const float* __restrict__ x,
                       const float* __restrict__ mr, const float* __restrict__ mw,
                       const float* __restrict__ mk, const float* __restrict__ mv,
                       const float* __restrict__ ma, const float* __restrict__ mg,
                       float* o0, float* o1, float* o2, float* o3, float* o4, float* o5) {
    size_t i = (size_t)blockIdx.x * blockDim.x + threadIdx.x;
    if (i >= (size_t)BTc * Cc) return;
    int c = (int)(i % Cc);
    size_t bt = i / Cc;
    int t = (int)(bt % Tc);
    float xv = x[i];
    float prev = (t == 0) ? 0.f : x[i - Cc];
    float xx = prev - xv;
    o0[i] = xv + xx * mr[c]; o1[i] = xv + xx * mw[c]; o2[i] = xv + xx * mk[c];
    o3[i] = xv + xx * mv[c]; o4[i] = xv + xx * ma[c]; o5[i] = xv + xx * mg[c];
}

__global__ void k_mix1(const float* __restrict__ x, const float* __restrict__ mk,
                       float* __restrict__ o) {
    size_t i = (size_t)blockIdx.x * blockDim.x + threadIdx.x;
    if (i >= (size_t)BTc * Cc) return;
    int c = (int)(i % Cc);
    size_t bt = i / Cc;
    int t = (int)(bt % Tc);
    float xv = x[i];
    float prev = (t == 0) ? 0.f : x[i - Cc];
    o[i] = xv + (prev - xv) * mk[c];
}

// ---------------- small elementwise kernels ----------------
__global__ void k_tanh(float* x, int n) {
    int i = blockIdx.x * blockDim.x + threadIdx.x;
    if (i < n) x[i] = tanhf(x[i]);
}
__global__ void k_sigmoid(float* x, int n) {
    int i = blockIdx.x * blockDim.x + threadIdx.x;
    if (i < n) x[i] = 1.f / (1.f + expf(-x[i]));
}
__global__ void k_relusq(float* x, int n) {
    int i = blockIdx.x * blockDim.x + threadIdx.x;
    if (i < n) { float t = fmaxf(x[i], 0.f); x[i] = t * t; }
}
__global__ void k_add(float* __restrict__ x, const float* __restrict__ t, int n) {
    int i = blockIdx.x * blockDim.x + threadIdx.x;
    if (i < n) x[i] += t[i];
}
// w = -softplus(-(w0 + lora)) - 0.5   (stable softplus)
__global__ void k_w_final(const float* __restrict__ w0, const float* __restrict__ l2,
                          float* __restrict__ w, int n) {
    int i = blockIdx.x * blockDim.x + threadIdx.x;
    if (i >= n) return;
    float y = w0[i % Cc] + l2[i];
    float z = -y;
    float sp = fmaxf(z, 0.f) + log1pf(expf(-fabsf(z)));
    w[i] = -sp - 0.5f;
}
__global__ void k_a_final(const float* __restrict__ a0, const float* __restrict__ l2,
                          float* __restrict__ a, int n) {
    int i = blockIdx.x * blockDim.x + threadIdx.x;
    if (i >= n) return;
    a[i] = 1.f / (1.f + expf(-(a0[i % Cc] + l2[i])));
}
// v = v + (v_first - v) * sigmoid(v0 + lora)
__global__ void k_v_update(const float* __restrict__ v0, const float* __restrict__ l2,
                           const float* __restrict__ vf, float* __restrict__ v, int n) {
    int i = blockIdx.x * blockDim.x + threadIdx.x;
    if (i >= n) return;
    float s = 1.f / (1.f + expf(-(v0[i % Cc] + l2[i])));
    v[i] += (vf[i] - v[i]) * s;
}

// ---------------- kk normalize (per head) + k2 ----------------
__global__ void k_kk_k2(const float* __restrict__ k, const float* __restrict__ a,
                        const float* __restrict__ kkw, const float* __restrict__ kaw,
                        float* __restrict__ kk, float* __restrict__ k2) {
    int th = blockIdx.x;              // BT*H blocks
    int h = th % Hc;
    int i = threadIdx.x;              // 0..63
    size_t base = (size_t)(th / Hc) * Cc + (size_t)h * HSc;
    __shared__ float s1[HSc];
    float kv = k[base + i];
    float t  = kv * kkw[h * HSc + i];
    s1[i] = t * t; __syncthreads();
    for (int off = 32; off > 0; off >>= 1) {
        if (i < off) s1[i] += s1[i + off];
        __syncthreads();
    }
    float nrm = fmaxf(sqrtf(s1[0]), 1e-12f);
    kk[base + i] = t / nrm;
    k2[base + i] = kv * (1.f + (a[base + i] - 1.f) * kaw[h * HSc + i]);
}

// ---------------- WKV7 recurrence: one block per (b,h), 64 threads ----------
__global__ void k_wkv7(const float* __restrict__ r, const float* __restrict__ wr,
                       const float* __restrict__ k, const float* __restrict__ v,
                       const float* __restrict__ kk, const float* __restrict__ a,
                       float* __restrict__ y) {
    int bh = blockIdx.x;
    int b = bh / Hc, h = bh % Hc;
    int i = threadIdx.x;
    __shared__ float sr[HSc], sw[HSc], sk[HSc], sv[HSc], sA[HSc], sB[HSc];
    float S[HSc];
#pragma unroll
    for (int j = 0; j < HSc; ++j) S[j] = 0.f;
    for (int t = 0; t < Tc; ++t) {
        size_t base = ((size_t)b * Tc + t) * Cc + (size_t)h * HSc;
        float rv = r[base + i];
        float wv = expf(-expf(wr[base + i]));
        float kv = k[base + i];
        float vv = v[base + i];
        float kkv = kk[base + i];
        float av  = a[base + i];
        __syncthreads();
        sr[i] = rv; sw[i] = wv; sk[i] = kv; sv[i] = vv;
        sA[i] = -kkv; sB[i] = kkv * av;
        __syncthreads();
        float sa = 0.f;
#pragma unroll
        for (int j = 0; j < HSc; ++j) sa += S[j] * sA[j];
        float vi = sv[i];
        float yv = 0.f;
#pragma unroll
        for (int j = 0; j < HSc; ++j) {
            float s = S[j] * sw[j] + sa * sB[j] + vi * sk[j];
            S[j] = s;
            yv += s * sr[j];
        }
        y[base + i] = yv;
    }
}

// ---------------- per-head GroupNorm + (r*k2*rk).sum * v bonus + gate ------
__global__ void k_postmix(const float* __restrict__ y, const float* __restrict__ gw,
                          const float* __restrict__ gb, const float* __restrict__ r,
                          const float* __restrict__ k2, const float* __restrict__ v,
                          const float* __restrict__ rk, const float* __restrict__ g,
                          float* __restrict__ out) {
    int th = blockIdx.x;
    int h = th % Hc;
    int i = threadIdx.x;
    size_t base = (size_t)(th / Hc) * Cc + (size_t)h * HSc;
    __shared__ float s1[HSc], s2[HSc];
    float yv = y[base + i];
    s1[i] = yv; s2[i] = yv * yv; __syncthreads();
    for (int off = 32; off > 0; off >>= 1) {
        if (i < off) { s1[i] += s1[i + off]; s2[i] += s2[i + off]; }
        __syncthreads();
    }
    float mean = s1[0] * (1.f / HSc);
    float var  = s2[0] * (1.f / HSc) - mean * mean;
    float yn = (yv - mean) * rsqrtf(var + 64e-5f) * gw[h * HSc + i] + gb[h * HSc + i];
    __syncthreads();
    s1[i] = r[base + i] * k2[base + i] * rk[h * HSc + i]; __syncthreads();
    for (int off = 32; off > 0; off >>= 1) {
        if (i < off) s1[i] += s1[i + off];
        __syncthreads();
    }
    out[base + i] = (yn + s1[0] * v[base + i]) * g[base + i];
}

// ---------------- head: sigmoid(x[b, T-1, :] @ head_w + head_b) ----------
__global__ void k_head(const float* __restrict__ x, const float* __restrict__ hw,
                       const float* __restrict__ hb, float* __restrict__ out) {
    int b = blockIdx.x, tid = threadIdx.x;
    __shared__ float s[256];
    const float* xp = x + ((size_t)b * Tc + (Tc - 1)) * Cc;
    float acc = 0.f;
    for (int c = tid; c < Cc; c += 256) acc += xp[c] * hw[c];
    s[tid] = acc; __syncthreads();
    for (int off = 128; off > 0; off >>= 1) {
        if (tid < off) s[tid] += s[tid + off];
        __syncthreads();
    }
    if (tid == 0) out[b] = 1.f / (1.f + expf(-(s[0] + hb[0])));
}

// ================= host side =================
static inline void gemm(const float* X, const float* W, float* Y,
                        int M, int N, int K, hipStream_t s) {
    if ((M & 63) == 0 && (N & 63) == 0 && (K & 31) == 0) {
        int blocks = (M / 64) * (N / 64);
        k_gemm_tiled<<<blocks, 256, 0, s>>>(X, W, Y, M, N, K);
    } else {
        int tiles = (M / 16) * (N / 16);
        k_gemm_f32<<<(tiles + 7) / 8, 256, 0, s>>>(X, W, Y, M, N, K);
    }
}
static inline int ceil_div(long long n, int d) { return (int)((n + d - 1) / d); }

extern "C" void kernel_launch(void* const* d_in, const int* in_sizes, int n_in,
                              void* d_out, int out_size, void* d_ws, size_t ws_size,
                              hipStream_t stream) {
    (void)out_size; (void)ws_size;
    // Locate idx by its unique element count (B*T=4096); params follow in dict order.
    int idx_pos = -1;
    for (int i = 0; i < n_in; ++i) if (in_sizes[i] == Bc * Tc) { idx_pos = i; break; }
    if (idx_pos < 0) idx_pos = 0;
    const float* P[40]; int pi = 0;
    for (int i = 0; i < n_in && pi < 40; ++i) {
        if (i == idx_pos) continue;
        P[pi++] = (const float*)d_in[i];
    }
    const int* idx = (const int*)d_in[idx_pos];
    enum { EMB=0, LN0W, LN0B, LN1W, LN1B, LN2W, LN2B,
           XR, XW, XK, XV, XA, XG, W0, W1, W2, A0, A1, A2, V0, V1, V2, G1, G2,
           KK, KA, RK, WR, WKm, WVm, WO, LNXW, LNXB, FXK, WKF, WVF,
           LNOW, LNOB, HW, HB };

    // workspace partition (floats)
    float* wsf = (float*)d_ws;
    size_t o = 0;
    const size_t NC = (size_t)BTc * Cc;
    float* bx    = wsf + o; o += NC;
    float* bxln  = wsf + o; o += NC;
    float* m[6]; for (int j = 0; j < 6; ++j) { m[j] = wsf + o; o += NC; }
    float* pr   = wsf + o; o += NC;
    float* pw   = wsf + o; o += NC;
    float* pk   = wsf + o; o += NC;
    float* pv   = wsf + o; o += NC;
    float* pa   = wsf + o; o += NC;
    float* pg   = wsf + o; o += NC;
    float* pkk  = wsf + o; o += NC;
    float* pk2  = wsf + o; o += NC;
    float* py   = wsf + o; o += NC;
    float* ptmp = wsf + o; o += NC;
    float* pvf  = wsf + o; o += NC;
    float* pl1  = wsf + o; o += (size_t)BTc * DGc;
    float* pl2  = wsf + o; o += NC;
    float* pffn = wsf + o; o += (size_t)BTc * Fc;

    const int nNC  = BTc * Cc;
    const int gNC  = ceil_div(nNC, 256);
    const int gNF  = ceil_div((long long)BTc * Fc, 256);

    // embedding + ln0
    k_embed<<<gNC, 256, 0, stream>>>(P[EMB], idx, ptmp);
    k_layernorm<<<BTc, 256, 0, stream>>>(ptmp, P[LN0W], P[LN0B], bx, 1e-5f);

    for (int i = 0; i < Lc; ++i) {
        const size_t iC = (size_t)i * Cc;
        // ---- time mix ----
        k_layernorm<<<BTc, 256, 0, stream>>>(bx, P[LN1W] + iC, P[LN1B] + iC, bxln, 1e-5f);
        k_mix6<<<gNC, 256, 0, stream>>>(bxln, P[XR] + iC, P[XW] + iC, P[XK] + iC,
                                        P[XV] + iC, P[XA] + iC, P[XG] + iC,
                                        m[0], m[1], m[2], m[3], m[4], m[5]);
        gemm(m[0], P[WR]  + (size_t)i * Cc * Cc, pr, BTc, Cc, Cc, stream);
        // w path: tanh(xw@w1)@w2 -> -softplus(-(w0+.)) - 0.5
        gemm(m[1], P[W1] + (size_t)i * Cc * DWc, pl1, BTc, DWc, Cc, stream);
        k_tanh<<<ceil_div(BTc * DWc, 256), 256, 0, stream>>>(pl1, BTc * DWc);
        gemm(pl1, P[W2] + (size_t)i * DWc * Cc, pl2, BTc, Cc, DWc, stream);
        k_w_final<<<gNC, 256, 0, stream>>>(P[W0] + iC, pl2, pw, nNC);
        gemm(m[2], P[WKm] + (size_t)i * Cc * Cc, pk, BTc, Cc, Cc, stream);
        gemm(m[3], P[WVm] + (size_t)i * Cc * Cc, pv, BTc, Cc, Cc, stream);
        if (i == 0) {
            hipMemcpyAsync(pvf, pv, NC * sizeof(float), hipMemcpyDeviceToDevice, stream);
        } else {
            gemm(m[3], P[V1] + (size_t)i * Cc * DVc, pl1, BTc, DVc, Cc, stream);
            gemm(pl1, P[V2] + (size_t)i * DVc * Cc, pl2, BTc, Cc, DVc, stream);
            k_v_update<<<gNC, 256, 0, stream>>>(P[V0] + iC, pl2, pvf, pv, nNC);
        }
        // a path
        gemm(m[4], P[A1] + (size_t)i * Cc * DAc, pl1, BTc, DAc, Cc, stream);
        gemm(pl1, P[A2] + (size_t)i * DAc * Cc, pl2, BTc, Cc, DAc, stream);
        k_a_final<<<gNC, 256, 0, stream>>>(P[A0] + iC, pl2, pa, nNC);
        // g path
        gemm(m[5], P[G1] + (size_t)i * Cc * DGc, pl1, BTc, DGc, Cc, stream);
        k_sigmoid<<<ceil_div(BTc * DGc, 256), 256, 0, stream>>>(pl1, BTc * DGc);
        gemm(pl1, P[G2] + (size_t)i * DGc * Cc, pg, BTc, Cc, DGc, stream);
        // kk / k2
        k_kk_k2<<<BTc * Hc, HSc, 0, stream>>>(pk, pa, P[KK] + iC, P[KA] + iC, pkk, pk2);
        // WKV7 scan
        k_wkv7<<<Bc * Hc, HSc, 0, stream>>>(pr, pw, pk2, pv, pkk, pa, py);
        // groupnorm + bonus + gate, then Wo and residual
        k_postmix<<<BTc * Hc, HSc, 0, stream>>>(py, P[LNXW] + iC, P[LNXB] + iC,
                                                pr, pk2, pv, P[RK] + iC, pg, m[0]);
        gemm(m[0], P[WO] + (size_t)i * Cc * Cc, ptmp, BTc, Cc, Cc, stream);
        k_add<<<gNC, 256, 0, stream>>>(bx, ptmp, nNC);
        // ---- channel mix ----
        k_layernorm<<<BTc, 256, 0, stream>>>(bx, P[LN2W] + iC, P[LN2B] + iC, bxln, 1e-5f);
        k_mix1<<<gNC, 256, 0, stream>>>(bxln, P[FXK] + iC, m[0]);
        gemm(m[0], P[WKF] + (size_t)i * Cc * Fc, pffn, BTc, Fc, Cc, stream);
        k_relusq<<<gNF, 256, 0, stream>>>(pffn, BTc * Fc);
        gemm(pffn, P[WVF] + (size_t)i * Fc * Cc, ptmp, BTc, Cc, Fc, stream);
        k_add<<<gNC, 256, 0, stream>>>(bx, ptmp, nNC);
    }

    // final LN + head
    k_layernorm<<<BTc, 256, 0, stream>>>(bx, P[LNOW], P[LNOB], bxln, 1e-5f);
    k_head<<<Bc, 256, 0, stream>>>(bxln, P[HW], P[HB], (float*)d_out);
}